// CustomModel_71725953843992
// MI455X (gfx1250) — compile-verified
//
#include <hip/hip_runtime.h>

typedef float v2f __attribute__((ext_vector_type(2)));
typedef float v8f __attribute__((ext_vector_type(8)));

#define RDIM 4096
#define DDIM 128
#define CDIM 8
#define TILE 64
#define KC   64
#define LSTR 68   // LDS row stride (floats): 16B-aligned rows, conflict-free b64 frag reads

__device__ __forceinline__ float wave_sum(float v) {
  #pragma unroll
  for (int off = 16; off > 0; off >>= 1) v += __shfl_xor(v, off, 32);
  return v;
}
__device__ __forceinline__ float wave_max(float v) {
  #pragma unroll
  for (int off = 16; off > 0; off >>= 1) v = fmaxf(v, __shfl_xor(v, off, 32));
  return v;
}

// scal layout (floats in d_ws):
// [0]=wsum_offdiag [1]=dmax_bits(u32) [2]=sumS [3]=pn [4]=pp [5]=sdiag2 [6]=sumW
__global__ void init_scal(float* scal) {
  if (threadIdx.x < 8) scal[threadIdx.x] = 0.0f;
}

__global__ __launch_bounds__(128)
void dist_tile_kernel(const float* __restrict__ feat, const float* __restrict__ S,
                      const float* __restrict__ W, float* __restrict__ scal) {
  __shared__ float lA[TILE * LSTR];
  __shared__ float lB[TILE * LSTR];
  __shared__ float lnA[TILE];
  __shared__ float lnB[TILE];

  const int tid  = threadIdx.x;
  const int wave = tid >> 5;
  const int lane = tid & 31;
  const int lo   = lane & 15;
  const int hi   = lane >> 4;
  const int iT   = blockIdx.y * TILE;
  const int jT   = blockIdx.x * TILE;

  const v8f vz = {0.f,0.f,0.f,0.f,0.f,0.f,0.f,0.f};
  v8f dist[4] = {vz, vz, vz, vz};

  for (int c = 0; c < CDIM; ++c) {
    v8f g[4] = {vz, vz, vz, vz};
    float npA = 0.0f, npB = 0.0f;
    const float* fbase = feat + (size_t)c * RDIM * DDIM;
    const float* wrow  = W + c * DDIM;

    for (int kc = 0; kc < DDIM; kc += KC) {
      __syncthreads();
      // Stage scaled A (rows iT..) and B (rows jT..) K-chunks into LDS.
      #pragma unroll
      for (int p = 0; p < 8; ++p) {
        int linear = p * 128 + tid;          // 1024 float4 units = 64x64 floats
        int row = linear >> 4;
        int c4  = (linear & 15) << 2;
        float4 wv = *(const float4*)(wrow + kc + c4);
        float4 av = *(const float4*)(fbase + (size_t)(iT + row) * DDIM + kc + c4);
        float4 bv = *(const float4*)(fbase + (size_t)(jT + row) * DDIM + kc + c4);
        av.x *= wv.x; av.y *= wv.y; av.z *= wv.z; av.w *= wv.w;
        bv.x *= wv.x; bv.y *= wv.y; bv.z *= wv.z; bv.w *= wv.w;
        *(float4*)&lA[row * LSTR + c4] = av;
        *(float4*)&lB[row * LSTR + c4] = bv;
      }
      __syncthreads();

      // Gram accumulation with f32 WMMA: G[m][n] += sum_k A[m][k]*B'[n][k]
      #pragma unroll
      for (int k0 = 0; k0 < KC; k0 += 4) {
        v2f a = *(const v2f*)&lA[(wave * 16 + lo) * LSTR + k0 + 2 * hi];
        #pragma unroll
        for (int n = 0; n < 4; ++n) {
          v2f b = *(const v2f*)&lB[(n * 16 + lo) * LSTR + k0 + 2 * hi];
          g[n] = __builtin_amdgcn_wmma_f32_16x16x4_f32(
              false, a, false, b, (short)0, g[n], false, false);
        }
      }

      // Row-norm partials from the staged (already W-scaled) tiles.
      {
        const float* src = (tid < 64) ? &lA[tid * LSTR] : &lB[(tid - 64) * LSTR];
        float acc = 0.0f;
        #pragma unroll
        for (int k = 0; k < KC; k += 4) {
          float4 v = *(const float4*)(src + k);
          acc += (v.x*v.x + v.y*v.y) + (v.z*v.z + v.w*v.w);
        }
        if (tid < 64) npA += acc; else npB += acc;
      }
    }

    if (tid < 64) lnA[tid] = npA; else lnB[tid - 64] = npB;
    __syncthreads();

    // Per-channel: dist += relu(n_i + n_j - 2*G)
    float nI[8];
    #pragma unroll
    for (int v = 0; v < 8; ++v) nI[v] = lnA[wave * 16 + v + 8 * hi];
    #pragma unroll
    for (int n = 0; n < 4; ++n) {
      float nJ = lnB[n * 16 + lo];
      #pragma unroll
      for (int v = 0; v < 8; ++v) {
        float d = nI[v] + nJ - 2.0f * g[n][v];
        dist[n][v] += fmaxf(d, 0.0f);
      }
    }
  }

  // Fused epilogue: off-diagonal S^2-weighted sum + global max.
  float wsum = 0.0f, dmax = 0.0f;
  #pragma unroll
  for (int n = 0; n < 4; ++n) {
    #pragma unroll
    for (int v = 0; v < 8; ++v) {
      int i = iT + wave * 16 + v + 8 * hi;
      int j = jT + n * 16 + lo;
      float dd = dist[n][v];
      dmax = fmaxf(dmax, dd);
      if (i != j) {
        float sv = S[(size_t)i * RDIM + j];
        wsum += dd * sv * sv;
      }
    }
  }
  wsum = wave_sum(wsum);
  dmax = wave_max(dmax);
  if (lane == 0) {
    atomicAdd(&scal[0], wsum);
    atomicMax((unsigned int*)scal + 1, __float_as_uint(dmax)); // dist >= 0 -> uint order ok
  }
}

__global__ __launch_bounds__(256)
void s_reduce_copy(const float* __restrict__ S, float* __restrict__ outS,
                   float* __restrict__ scal) {
  size_t idx    = (size_t)blockIdx.x * 256 + threadIdx.x;
  size_t stride = (size_t)gridDim.x * 256;
  const float4* S4 = (const float4*)S;
  const size_t n4 = (size_t)RDIM * RDIM / 4;
  float sum = 0.f, pn = 0.f, pp = 0.f;
  for (size_t i = idx; i < n4; i += stride) {
    float4 v = S4[i];
    sum += (v.x + v.y) + (v.z + v.w);
    pn  += fminf(v.x, 0.f) + fminf(v.y, 0.f) + fminf(v.z, 0.f) + fminf(v.w, 0.f);
    pp  += fmaxf(v.x - 1.f, 0.f) + fmaxf(v.y - 1.f, 0.f) +
           fmaxf(v.z - 1.f, 0.f) + fmaxf(v.w - 1.f, 0.f);
    size_t o = 4 * i;                 // outS is only 4B-aligned (d_out+1)
    outS[o] = v.x; outS[o+1] = v.y; outS[o+2] = v.z; outS[o+3] = v.w;
  }
  sum = wave_sum(sum); pn = wave_sum(pn); pp = wave_sum(pp);
  if ((threadIdx.x & 31) == 0) {
    atomicAdd(&scal[2], sum);
    atomicAdd(&scal[3], pn);
    atomicAdd(&scal[4], pp);
  }
}

__global__ __launch_bounds__(256)
void w_reduce_copy(const float* __restrict__ W, float* __restrict__ outW,
                   float* __restrict__ scal) {
  float sum = 0.f;
  for (int i = threadIdx.x; i < CDIM * DDIM; i += 256) {
    float v = W[i];
    outW[i] = v;
    sum += v;
  }
  sum = wave_sum(sum);
  if ((threadIdx.x & 31) == 0) atomicAdd(&scal[6], sum);
}

__global__ __launch_bounds__(256)
void diag_kernel(const float* __restrict__ S, float* __restrict__ scal) {
  float acc = 0.f;
  for (int i = threadIdx.x; i < RDIM; i += 256) {
    float s = S[(size_t)i * RDIM + i];
    acc += s * s;
  }
  acc = wave_sum(acc);
  if ((threadIdx.x & 31) == 0) atomicAdd(&scal[5], acc);
}

__global__ void finalize(const float* __restrict__ scal, float* __restrict__ out) {
  float wsum   = scal[0];
  float dmax   = __uint_as_float(((const unsigned int*)scal)[1]);
  float sumS   = scal[2];
  float pn     = scal[3];
  float pp     = scal[4];
  float sdiag2 = scal[5];
  float sumW   = scal[6];
  float dist_S = wsum + dmax * sdiag2;
  // BETA = 1; penalty_sum1 terms collapse to |total - count|
  float loss = dist_S + ((-pn) + pp)
             + 100.0f * fabsf(sumW - (float)DDIM)
             + 100.0f * fabsf(sumS - (float)RDIM);
  out[0] = loss;
}

extern "C" void kernel_launch(void* const* d_in, const int* in_sizes, int n_in,
                              void* d_out, int out_size, void* d_ws, size_t ws_size,
                              hipStream_t stream) {
  const float* feat = (const float*)d_in[0];   // (8, 4096, 128)
  const float* S    = (const float*)d_in[1];   // (4096, 4096)
  const float* W    = (const float*)d_in[2];   // (8, 1, 128)
  float* out  = (float*)d_out;                 // [loss, S(16M), W(1024)]
  float* outS = out + 1;
  float* outW = out + 1 + (size_t)RDIM * RDIM;
  float* scal = (float*)d_ws;

  init_scal<<<1, 32, 0, stream>>>(scal);

  dim3 grid(RDIM / TILE, RDIM / TILE);
  dist_tile_kernel<<<grid, 128, 0, stream>>>(feat, S, W, scal);

  s_reduce_copy<<<4096, 256, 0, stream>>>(S, outS, scal);
  w_reduce_copy<<<1, 256, 0, stream>>>(W, outW, scal);
  diag_kernel<<<1, 256, 0, stream>>>(S, scal);

  finalize<<<1, 1, 0, stream>>>(scal, out);
}